// SMFWPlayer_80805514707489
// MI455X (gfx1250) — compile-verified
//
#include <hip/hip_runtime.h>
#include <math.h>

// ---- problem constants (from reference) ----
#define TT  1024
#define BB  32
#define HH  8
#define DD  32
#define YD  97          // 3*D + 1
#define INN 256         // H*D
#define ETOT (YD + DD + DD + 4)   // 165: [Wy | Wq | Wk | wb] concatenated columns

typedef __attribute__((ext_vector_type(2))) float v2f;
typedef __attribute__((ext_vector_type(8))) float v8f;

// ---------------------------------------------------------------------------
// Kernel 1: self-referential weight-matrix scan + delta-rule fast-weight memory.
// One workgroup (256 threads = 8 wave32) per (b, head) chain; all recurrent
// state lives in LDS for the entire T=1024 scan. Writes fout (T*B, 256) to ws.
// ---------------------------------------------------------------------------
__global__ __launch_bounds__(256)
void srwm_scan_kernel(const float* __restrict__ hin,
                      const float* __restrict__ W_y,
                      const float* __restrict__ W_q,
                      const float* __restrict__ W_k,
                      const float* __restrict__ w_b,
                      const float* __restrict__ state_W_y,
                      const float* __restrict__ state_W_q,
                      const float* __restrict__ state_W_k,
                      const float* __restrict__ state_w_b,
                      const float* __restrict__ fastw0,
                      float* __restrict__ fout)
{
    __shared__ float sM[DD * ETOT];   // concatenated [Wy|Wq|Wk|wb], row d, col e
    __shared__ float sF[DD * DD];     // fast-weight matrix
    __shared__ float sx[DD];          // softmax(x_t)
    __shared__ float sy[ETOT];        // y_all = x_t @ Mcat
    __shared__ float svq[ETOT];
    __shared__ float svk[ETOT];
    __shared__ float svold[DD];
    __shared__ float sq[DD], sk[DD], sfq[DD], sfk[DD];
    __shared__ float sbeta[4];
    __shared__ float sfb;

    const int tid  = threadIdx.x;
    const int lane = tid & 31;
    const int wv   = tid >> 5;
    const int b    = blockIdx.x / HH;
    const int head = blockIdx.x % HH;
    const int bh   = b * HH + head;

    // ---- initialize state: state_W_* + W_* (W_* broadcast over batch) ----
    for (int idx = tid; idx < DD * ETOT; idx += 256) {
        const int d = idx / ETOT;
        const int e = idx % ETOT;
        float v;
        if (e < YD) {
            v = state_W_y[(bh * DD + d) * YD + e] + W_y[(head * DD + d) * YD + e];
        } else if (e < YD + DD) {
            const int c = e - YD;
            v = state_W_q[(bh * DD + d) * DD + c] + W_q[(head * DD + d) * DD + c];
        } else if (e < YD + 2 * DD) {
            const int c = e - YD - DD;
            v = state_W_k[(bh * DD + d) * DD + c] + W_k[(head * DD + d) * DD + c];
        } else {
            const int c = e - YD - 2 * DD;
            v = state_w_b[(bh * DD + d) * 4 + c] + w_b[(head * DD + d) * 4 + c];
        }
        sM[idx] = v;
    }
    for (int idx = tid; idx < DD * DD; idx += 256)
        sF[idx] = fastw0[bh * DD * DD + idx];
    __syncthreads();

    for (int t = 0; t < TT; ++t) {
        // ---- phase 1: x_t = softmax(h[t,b,head,:]) (wave 0) ----
        if (wv == 0) {
            float xv = hin[(t * BB + b) * INN + head * DD + lane];
            float m = xv;
            #pragma unroll
            for (int off = 16; off > 0; off >>= 1) m = fmaxf(m, __shfl_xor(m, off, 32));
            float ex = expf(xv - m);
            float s = ex;
            #pragma unroll
            for (int off = 16; off > 0; off >>= 1) s += __shfl_xor(s, off, 32);
            sx[lane] = ex / s;
        }
        __syncthreads();

        // ---- phase 2: y_all[e] = sum_d x[d] * Mcat[d,e] ----
        if (tid < ETOT) {
            float acc = 0.f;
            #pragma unroll
            for (int d = 0; d < DD; ++d) acc += sx[d] * sM[d * ETOT + tid];
            sy[tid] = acc;
        }
        __syncthreads();

        // ---- phase 3: activations (one wave each) ----
        if (wv < 4) {
            const float* src = (wv == 0) ? &sy[YD]
                             : (wv == 1) ? &sy[YD + DD]
                             : (wv == 2) ? &sy[0]
                                         : &sy[DD];
            float* dst = (wv == 0) ? sq : (wv == 1) ? sk : (wv == 2) ? sfq : sfk;
            float v = src[lane];
            float m = v;
            #pragma unroll
            for (int off = 16; off > 0; off >>= 1) m = fmaxf(m, __shfl_xor(m, off, 32));
            float ex = expf(v - m);
            float s = ex;
            #pragma unroll
            for (int off = 16; off > 0; off >>= 1) s += __shfl_xor(s, off, 32);
            dst[lane] = ex / s;
        } else if (wv == 4) {
            if (lane < 4)      sbeta[lane] = 1.f / (1.f + expf(-sy[YD + 2 * DD + lane]));
            else if (lane == 4) sfb = 1.f / (1.f + expf(-sy[3 * DD]));   // y[96]
        }
        __syncthreads();

        // ---- phase 4: vq/vk over all columns; v_old for fast weights ----
        if (tid < ETOT) {
            float aq = 0.f, ak = 0.f;
            #pragma unroll
            for (int d = 0; d < DD; ++d) {
                const float mv = sM[d * ETOT + tid];
                aq += sq[d] * mv;
                ak += sk[d] * mv;
            }
            svq[tid] = aq;
            svk[tid] = ak;
        } else if (wv == 6) {   // tid in [192,224)
            float a = 0.f;
            #pragma unroll
            for (int d = 0; d < DD; ++d) a += sfk[d] * sF[d * DD + lane];
            svold[lane] = a;
        }
        __syncthreads();

        // ---- phase 5: rank-1 updates of Mcat and F ----
        for (int idx = tid; idx < DD * ETOT; idx += 256) {
            const int d = idx / ETOT;
            const int e = idx % ETOT;
            const float bta = (e < YD)          ? sbeta[0]
                            : (e < YD + DD)     ? sbeta[1]
                            : (e < YD + 2 * DD) ? sbeta[2]
                                                : sbeta[3];
            sM[idx] += bta * sk[d] * (svq[e] - svk[e]);
        }
        for (int idx = tid; idx < DD * DD; idx += 256) {
            const int d = idx >> 5;
            const int e = idx & 31;
            sF[idx] += sfb * sfk[d] * (sy[2 * DD + e] - svold[e]);  // fv = y[64:96]
        }
        __syncthreads();

        // ---- phase 6: out_t = fq @ F_new, write to workspace ----
        if (wv == 0) {
            float a = 0.f;
            #pragma unroll
            for (int d = 0; d < DD; ++d) a += sfq[d] * sF[d * DD + lane];
            fout[(t * BB + b) * INN + head * DD + lane] = a;
        }
        // next iteration's phase-1 barrier orders phase 6 reads vs. later writes
    }
}

// ---------------------------------------------------------------------------
// Kernel 2: d_out = h + fout @ W_out^T  via fp32 WMMA (V_WMMA_F32_16X16X4_F32).
// One 16x16 C tile per wave, K stepped by 4. M=32768, N=256, K=256.
// A/B fragment lane layout: element (lane%16, K = vgpr + 2*(lane/16)).
// ---------------------------------------------------------------------------
__global__ __launch_bounds__(256)
void out_gemm_wmma_kernel(const float* __restrict__ fout,
                          const float* __restrict__ Wout,
                          const float* __restrict__ hin,
                          float* __restrict__ out)
{
    const int waveId = blockIdx.x * 8 + (threadIdx.x >> 5);
    const int tileN  = waveId & 15;        // 256/16 = 16 N tiles
    const int tileM  = waveId >> 4;        // 32768/16 = 2048 M tiles
    const int lane   = threadIdx.x & 31;
    const int r      = lane & 15;          // row (A) / col (B) within tile
    const int kh     = lane >> 4;          // K half select

    const float* Arow = fout + (size_t)(tileM * 16 + r) * INN;
    const float* Brow = Wout + (size_t)(tileN * 16 + r) * INN;

    v8f acc = {};
    #pragma unroll 4
    for (int k0 = 0; k0 < INN; k0 += 4) {
        const float2 av = *(const float2*)(Arow + k0 + 2 * kh);
        const float2 bv = *(const float2*)(Brow + k0 + 2 * kh);
        v2f a; a.x = av.x; a.y = av.y;
        v2f bb; bb.x = bv.x; bb.y = bv.y;
        acc = __builtin_amdgcn_wmma_f32_16x16x4_f32(
                  /*neg_a=*/false, a, /*neg_b=*/false, bb,
                  /*c_mod=*/(short)0, acc, /*reuse_a=*/false, /*reuse_b=*/false);
    }

    const int col = tileN * 16 + r;
    #pragma unroll
    for (int i = 0; i < 8; ++i) {
        const int row = tileM * 16 + i + 8 * kh;   // C layout: VGPR i -> M = i + 8*(lane/16)
        const size_t off = (size_t)row * INN + col;
        out[off] = hin[off] + acc[i];
    }
}

// ---------------------------------------------------------------------------
extern "C" void kernel_launch(void* const* d_in, const int* in_sizes, int n_in,
                              void* d_out, int out_size, void* d_ws, size_t ws_size,
                              hipStream_t stream) {
    const float* hin        = (const float*)d_in[0];
    const float* W_y        = (const float*)d_in[1];
    const float* W_q        = (const float*)d_in[2];
    const float* W_k        = (const float*)d_in[3];
    const float* w_b        = (const float*)d_in[4];
    const float* W_out      = (const float*)d_in[5];
    const float* state_W_y  = (const float*)d_in[6];
    const float* state_W_q  = (const float*)d_in[7];
    const float* state_W_k  = (const float*)d_in[8];
    const float* state_w_b  = (const float*)d_in[9];
    const float* fastw0     = (const float*)d_in[10];
    float* out  = (float*)d_out;
    float* fout = (float*)d_ws;          // (T*B, 256) fp32 intermediate = 32 MB

    // Phase 1: 256 chains, one workgroup each.
    srwm_scan_kernel<<<dim3(BB * HH), dim3(256), 0, stream>>>(
        hin, W_y, W_q, W_k, w_b,
        state_W_y, state_W_q, state_W_k, state_w_b, fastw0, fout);

    // Phase 2: (2048 M-tiles x 16 N-tiles) / 8 waves per block = 4096 blocks.
    out_gemm_wmma_kernel<<<dim3((TT * BB / 16) * (INN / 16) / 8), dim3(256), 0, stream>>>(
        fout, W_out, hin, out);
}